// KM3DHead_31980326486026
// MI455X (gfx1250) — compile-verified
//
#include <hip/hip_runtime.h>
#include <math.h>
#include <stdint.h>

#define B_    16
#define H_    96
#define W_    320
#define HW_   (H_*W_)      // 30720
#define CHM   3
#define J_    9
#define KTOP  100
#define NCOLS 45
#define THRESH_ 0.1f

typedef __attribute__((ext_vector_type(2))) float v2f;
typedef __attribute__((ext_vector_type(8))) float v8f;

// ---------------------------------------------------------------------------
// Kernel 1: per-(b,c) plane: async-copy plane HBM->LDS (no VGPR round trip),
// then sigmoid -> 3x3 NMS -> iterative top-K entirely in LDS.
// One workgroup (1024 threads = 32 waves) per plane; 128 KB dynamic LDS.
// ---------------------------------------------------------------------------
extern "C" __global__ __launch_bounds__(1024)
void k_nms_topk(const float* __restrict__ in, float* __restrict__ scoreOut,
                int* __restrict__ indOut)
{
    extern __shared__ char smem[];
    float* plane = (float*)smem;                        // HW_ floats
    float* rval  = (float*)(smem + (size_t)HW_*4);      // 1024 floats
    int*   ridx  = (int*)  (smem + (size_t)HW_*4 + 4096);
    const int tid = threadIdx.x;
    const int pid = blockIdx.x;                         // b*C + c
    const float* src = in + (size_t)pid * HW_;

    // --- async HBM -> LDS copy: one b128 per lane per iteration ------------
    const unsigned lds_base = (unsigned)(uintptr_t)smem;   // LDS byte offset
    for (int f4 = tid; f4 < HW_/4; f4 += 1024) {
        unsigned ldsb = lds_base + (unsigned)f4 * 16u;
        const float* g = src + (size_t)f4 * 4;
        asm volatile("global_load_async_to_lds_b128 %0, %1, off"
                     :: "v"(ldsb), "v"(g) : "memory");
    }
    asm volatile("s_wait_asynccnt 0x0" ::: "memory");
    __syncthreads();

    // --- sigmoid in place --------------------------------------------------
    #pragma unroll
    for (int i = 0; i < 30; ++i) {
        int idx = tid + i*1024;
        float x = plane[idx];
        plane[idx] = 1.0f / (1.0f + expf(-x));
    }
    __syncthreads();

    // --- 3x3 max window (SAME, -inf pad == skip OOB neighbors) -------------
    float loc[30];
    #pragma unroll
    for (int i = 0; i < 30; ++i) {
        int idx = tid + i*1024;
        int y = idx / W_, x = idx % W_;
        float c = plane[idx];
        float m = c;
        for (int dy = -1; dy <= 1; ++dy) {
            int yy = y + dy; if (yy < 0 || yy >= H_) continue;
            for (int dx = -1; dx <= 1; ++dx) {
                int xx = x + dx; if (xx < 0 || xx >= W_) continue;
                m = fmaxf(m, plane[yy*W_ + xx]);
            }
        }
        loc[i] = (c == m) ? c : 0.0f;
    }
    __syncthreads();
    #pragma unroll
    for (int i = 0; i < 30; ++i) plane[tid + i*1024] = loc[i];
    __syncthreads();

    // --- iterative top-K, stable (lowest-index) tie-break like lax.top_k ---
    for (int k = 0; k < KTOP; ++k) {
        float best = -INFINITY; int bidx = 0x7fffffff;
        #pragma unroll
        for (int i = 0; i < 30; ++i) {
            int idx = tid + i*1024; float v = plane[idx];
            if (v > best || (v == best && idx < bidx)) { best = v; bidx = idx; }
        }
        rval[tid] = best; ridx[tid] = bidx;
        __syncthreads();
        for (int s = 512; s > 0; s >>= 1) {
            if (tid < s) {
                float ov = rval[tid+s]; int oi = ridx[tid+s];
                if (ov > rval[tid] || (ov == rval[tid] && oi < ridx[tid])) {
                    rval[tid] = ov; ridx[tid] = oi;
                }
            }
            __syncthreads();
        }
        if (tid == 0) {
            scoreOut[pid*KTOP + k] = rval[0];
            indOut  [pid*KTOP + k] = ridx[0];
            plane[ridx[0]] = -INFINITY;
        }
        __syncthreads();
    }
}

// ---------------------------------------------------------------------------
// Kernel 2: per-batch second-level top-K over the 3*K=300 channel candidates.
// ---------------------------------------------------------------------------
extern "C" __global__ __launch_bounds__(256)
void k_topk2(const float* __restrict__ chScore, const int* __restrict__ chInd,
             float* __restrict__ scores, int* __restrict__ inds,
             int* __restrict__ clses, float* __restrict__ xs0,
             float* __restrict__ ys0)
{
    __shared__ float cv[CHM*KTOP];
    __shared__ float rv[256];
    __shared__ int   ri[256];
    const int tid = threadIdx.x, b = blockIdx.x;
    const int N = CHM*KTOP;
    for (int t = tid; t < N; t += 256) cv[t] = chScore[b*N + t];
    __syncthreads();
    for (int k = 0; k < KTOP; ++k) {
        float best = -INFINITY; int bidx = 0x7fffffff;
        for (int t = tid; t < N; t += 256) {
            float v = cv[t];
            if (v > best || (v == best && t < bidx)) { best = v; bidx = t; }
        }
        rv[tid] = best; ri[tid] = bidx;
        __syncthreads();
        for (int s = 128; s > 0; s >>= 1) {
            if (tid < s) {
                float ov = rv[tid+s]; int oi = ri[tid+s];
                if (ov > rv[tid] || (ov == rv[tid] && oi < ri[tid])) {
                    rv[tid] = ov; ri[tid] = oi;
                }
            }
            __syncthreads();
        }
        if (tid == 0) {
            int i2 = ri[0];
            int ind = chInd[b*N + i2];
            scores[b*KTOP + k] = rv[0];
            inds  [b*KTOP + k] = ind;
            clses [b*KTOP + k] = i2 / KTOP;
            ys0   [b*KTOP + k] = (float)(ind / W_);
            xs0   [b*KTOP + k] = (float)(ind % W_);
            cv[i2] = -INFINITY;
        }
        __syncthreads();
    }
}

// ---------------------------------------------------------------------------
// Kernel 3: gather features at top-K indices; write output cols 0-4,23-25,
// 35-44; stage unscaled bboxes and kps (already in (b,J,K,2) layout).
// ---------------------------------------------------------------------------
extern "C" __global__ void k_gather(
    const float* __restrict__ wh, const float* __restrict__ hps,
    const float* __restrict__ dimf, const float* __restrict__ rot,
    const float* __restrict__ prob, const float* __restrict__ reg,
    const float* __restrict__ scores, const int* __restrict__ inds,
    const int* __restrict__ clses, const float* __restrict__ xs0,
    const float* __restrict__ ys0,
    float* __restrict__ out, float* __restrict__ bboxRaw,
    float* __restrict__ kpsJK)
{
    int t = blockIdx.x*blockDim.x + threadIdx.x;
    if (t >= B_*KTOP) return;
    int b = t / KTOP, k = t % KTOP;
    int ind = inds[t];
    float x0 = xs0[t], y0 = ys0[t];
    float xs = x0 + reg[((size_t)b*2 + 0)*HW_ + ind];
    float ys = y0 + reg[((size_t)b*2 + 1)*HW_ + ind];
    float w0 = wh[((size_t)b*2 + 0)*HW_ + ind];
    float w1 = wh[((size_t)b*2 + 1)*HW_ + ind];
    float l = xs - w0*0.5f, tp = ys - w1*0.5f;
    float r = xs + w0*0.5f, bt = ys + w1*0.5f;

    float* det = out + (size_t)t*NCOLS;
    det[0] = l*4.f; det[1] = tp*4.f; det[2] = r*4.f; det[3] = bt*4.f;
    det[4] = scores[t];
    bboxRaw[t*4+0] = l; bboxRaw[t*4+1] = tp; bboxRaw[t*4+2] = r; bboxRaw[t*4+3] = bt;

    for (int j = 0; j < J_; ++j) {
        float kx = hps[((size_t)b*18 + 2*j    )*HW_ + ind] + x0;
        float ky = hps[((size_t)b*18 + 2*j + 1)*HW_ + ind] + y0;
        kpsJK[(((size_t)b*J_ + j)*KTOP + k)*2 + 0] = kx;
        kpsJK[(((size_t)b*J_ + j)*KTOP + k)*2 + 1] = ky;
    }
    det[23] = dimf[((size_t)b*3 + 0)*HW_ + ind];
    det[24] = dimf[((size_t)b*3 + 1)*HW_ + ind];
    det[25] = dimf[((size_t)b*3 + 2)*HW_ + ind];
    for (int c = 0; c < 8; ++c) det[35+c] = rot[((size_t)b*8 + c)*HW_ + ind];
    det[43] = prob[(size_t)b*HW_ + ind];
    det[44] = (float)clses[t];
}

// ---------------------------------------------------------------------------
// Kernel 4: hp_offset gather + validity masking of heatmap keypoint peaks.
// ---------------------------------------------------------------------------
extern "C" __global__ void k_hpadj(
    const float* __restrict__ hpScore, const int* __restrict__ hpInd,
    const float* __restrict__ hpOff,
    float* __restrict__ hmKps, float* __restrict__ hmScoreAdj)
{
    int t = blockIdx.x*blockDim.x + threadIdx.x;
    if (t >= B_*J_*KTOP) return;
    int b = t / (J_*KTOP);
    int ind = hpInd[t];
    float s  = hpScore[t];
    float hx = (float)(ind % W_) + hpOff[((size_t)b*2 + 0)*HW_ + ind];
    float hy = (float)(ind / W_) + hpOff[((size_t)b*2 + 1)*HW_ + ind];
    bool valid = s > THRESH_;
    hmScoreAdj[t] = valid ? s  : -1.0f;
    hmKps[t*2+0]  = valid ? hx : -10000.0f;
    hmKps[t*2+1]  = valid ? hy : -10000.0f;
}

// ---------------------------------------------------------------------------
// Kernel 5: WMMA keypoint association. One wave per (b,j).
// dist2[i][c] = |k_i|^2 + |h_c|^2 - 2 k.h as a single 16x16x4 f32 WMMA per
// tile: A row = [kx, ky, |k|^2, 1], B col = [-2hx, -2hy, 1, |h|^2].
// Operands are pre-packed in LDS in the per-lane WMMA register layout
// (A: M=lane%16, K-pair selected by lane half; B: N=lane%16, K-pair by half)
// so each operand fill is one branchless ds_load_b64 — no exec-mask churn.
// B tiles are RI-invariant and hoisted into registers.
// ---------------------------------------------------------------------------
extern "C" __global__ __launch_bounds__(32)
void k_assoc(const float* __restrict__ kpsJK, const float* __restrict__ hmKps,
             const float* __restrict__ hmScoreAdj,
             const float* __restrict__ bboxRaw, float* __restrict__ out)
{
    __shared__ __align__(16) float Apk[2][112][2]; // [half][M][pair]
    __shared__ __align__(16) float Bpk[2][112][2]; // [half][N][pair]
    __shared__ float Hx[112], Hy[112], Hsc[112];
    const int bj = blockIdx.x;            // b*J + j
    const int b = bj / J_, j = bj % J_;
    const int lane = threadIdx.x;

    for (int t = lane; t < 112; t += 32) {
        float kx = 0.f, ky = 0.f;
        if (t < KTOP) {
            kx = kpsJK[((size_t)bj*KTOP + t)*2 + 0];
            ky = kpsJK[((size_t)bj*KTOP + t)*2 + 1];
        }
        Apk[0][t][0] = kx;            Apk[0][t][1] = ky;     // K=0,1
        Apk[1][t][0] = kx*kx + ky*ky; Apk[1][t][1] = 1.0f;   // K=2,3
        float hx = 0.f, hy = 0.f, hn = 1e30f, hs = 0.f;      // pad cols: huge
        if (t < KTOP) {
            hx = hmKps[((size_t)bj*KTOP + t)*2 + 0];
            hy = hmKps[((size_t)bj*KTOP + t)*2 + 1];
            hn = hx*hx + hy*hy;
            hs = hmScoreAdj[(size_t)bj*KTOP + t];
        }
        Hx[t] = hx; Hy[t] = hy; Hsc[t] = hs;
        Bpk[0][t][0] = -2.f*hx; Bpk[0][t][1] = -2.f*hy;      // K=0,1
        Bpk[1][t][0] = 1.0f;    Bpk[1][t][1] = hn;           // K=2,3
    }
    __syncthreads();

    const int half = lane >> 4;
    const int lid  = lane & 15;
    const v2f* Abase = (const v2f*)(&Apk[half][0][0]);
    const v2f* Bbase = (const v2f*)(&Bpk[half][0][0]);

    v2f btile[7];
    #pragma unroll
    for (int CI = 0; CI < 7; ++CI) btile[CI] = Bbase[CI*16 + lid];

    for (int RI = 0; RI < 7; ++RI) {
        int rbase = RI*16;
        v2f a = Abase[rbase + lid];
        float minv[8]; int minc[8];
        #pragma unroll
        for (int v = 0; v < 8; ++v) { minv[v] = INFINITY; minc[v] = 0x7fffffff; }

        #pragma unroll
        for (int CI = 0; CI < 7; ++CI) {
            v8f cz = {};
            v8f d = __builtin_amdgcn_wmma_f32_16x16x4_f32(
                false, a, false, btile[CI], (short)0, cz, false, false);
            int col = CI*16 + lid;
            #pragma unroll
            for (int v = 0; v < 8; ++v) {
                float dv = d[v];
                if (dv < minv[v] || (dv == minv[v] && col < minc[v])) {
                    minv[v] = dv; minc[v] = col;
                }
            }
        }
        // reduce over the 16 lanes of each half (same row, 16 columns)
        #pragma unroll
        for (int m = 1; m < 16; m <<= 1) {
            #pragma unroll
            for (int v = 0; v < 8; ++v) {
                float ov = __shfl_xor(minv[v], m, 32);
                int   oc = __shfl_xor(minc[v], m, 32);
                if (ov < minv[v] || (ov == minv[v] && oc < minc[v])) {
                    minv[v] = ov; minc[v] = oc;
                }
            }
        }
        if (lid == 0) {
            #pragma unroll
            for (int v = 0; v < 8; ++v) {
                int row = rbase + v + half*8;
                if (row < KTOP) {
                    float md  = sqrtf(fmaxf(minv[v], 0.f));
                    int   cs  = minc[v];
                    float ssel = Hsc[cs], hxs = Hx[cs], hys = Hy[cs];
                    const float* bbp = bboxRaw + ((size_t)b*KTOP + row)*4;
                    float l = bbp[0], tp = bbp[1], r = bbp[2], bt = bbp[3];
                    bool inval = (hxs < l) || (hxs > r) || (hys < tp) || (hys > bt)
                              || (ssel < THRESH_)
                              || (md > fmaxf(bt - tp, r - l)*0.3f);
                    float ox = inval ? Apk[0][row][0] : hxs;
                    float oy = inval ? Apk[0][row][1] : hys;
                    float* det = out + ((size_t)b*KTOP + row)*NCOLS;
                    det[5 + 2*j    ] = ox*4.f;
                    det[5 + 2*j + 1] = oy*4.f;
                    det[26 + j     ] = ssel;
                }
            }
        }
    }
}

// ---------------------------------------------------------------------------
extern "C" void kernel_launch(void* const* d_in, const int* in_sizes, int n_in,
                              void* d_out, int out_size, void* d_ws, size_t ws_size,
                              hipStream_t stream)
{
    const float* hm     = (const float*)d_in[0];
    const float* wh     = (const float*)d_in[1];
    const float* hps    = (const float*)d_in[2];
    const float* dimf   = (const float*)d_in[3];
    const float* rot    = (const float*)d_in[4];
    const float* prob   = (const float*)d_in[5];
    const float* reg    = (const float*)d_in[6];
    const float* hm_hp  = (const float*)d_in[7];
    const float* hp_off = (const float*)d_in[8];
    float* out = (float*)d_out;

    char* ws = (char*)d_ws;
    size_t off = 0;
    auto alloc = [&](size_t bytes) -> char* {
        char* p = ws + off; off += (bytes + 255) & ~(size_t)255; return p;
    };
    float* hmScore = (float*)alloc((size_t)B_*CHM*KTOP*4);
    int*   hmInd   = (int*)  alloc((size_t)B_*CHM*KTOP*4);
    float* hpScore = (float*)alloc((size_t)B_*J_*KTOP*4);
    int*   hpInd   = (int*)  alloc((size_t)B_*J_*KTOP*4);
    float* scores  = (float*)alloc((size_t)B_*KTOP*4);
    int*   inds    = (int*)  alloc((size_t)B_*KTOP*4);
    int*   clses   = (int*)  alloc((size_t)B_*KTOP*4);
    float* xs0     = (float*)alloc((size_t)B_*KTOP*4);
    float* ys0     = (float*)alloc((size_t)B_*KTOP*4);
    float* bboxRaw = (float*)alloc((size_t)B_*KTOP*4*4);
    float* kpsJK   = (float*)alloc((size_t)B_*J_*KTOP*2*4);
    float* hmKps   = (float*)alloc((size_t)B_*J_*KTOP*2*4);
    float* hmScA   = (float*)alloc((size_t)B_*J_*KTOP*4);

    const size_t lds1 = (size_t)HW_*4 + 1024*4 + 1024*4;   // 128 KB < 320 KB WGP LDS
    k_nms_topk<<<B_*CHM, 1024, lds1, stream>>>(hm,    hmScore, hmInd);
    k_nms_topk<<<B_*J_,  1024, lds1, stream>>>(hm_hp, hpScore, hpInd);
    k_topk2<<<B_, 256, 0, stream>>>(hmScore, hmInd, scores, inds, clses, xs0, ys0);
    k_gather<<<(B_*KTOP + 255)/256, 256, 0, stream>>>(
        wh, hps, dimf, rot, prob, reg, scores, inds, clses, xs0, ys0,
        out, bboxRaw, kpsJK);
    k_hpadj<<<(B_*J_*KTOP + 255)/256, 256, 0, stream>>>(
        hpScore, hpInd, hp_off, hmKps, hmScA);
    k_assoc<<<B_*J_, 32, 0, stream>>>(kpsJK, hmKps, hmScA, bboxRaw, out);
}